// HigherOrderMessagePassing_36842229465189
// MI455X (gfx1250) — compile-verified
//
#include <hip/hip_runtime.h>
#include <stdint.h>

#define D_FEAT 64
#define TILE 32            // edges staged per wave per iteration
#define WAVES_PER_BLOCK 8
#define BLOCK 256
#define MAX_BLOCKS 1536

// ---------------------------------------------------------------------------
// Gather-scale-scatter kernel (placed FIRST so the disasm snippet shows it).
//  * Each wave32 owns a private LDS slice with 2 staging buffers of 32 edge
//    records (value, src, tgt), filled with global_load_async_to_lds_b32
//    (non-temporal: edge stream is touched once) and double-buffered across
//    tiles (ASYNCcnt-tracked, waits via s_wait_asynccnt).
//  * 16 lanes cooperate on one edge: lane loads one float4 of x[src]
//    (coalesced 256B/group global_load_b128, RT so x stays L2-resident),
//    scales by the edge value, and issues 4 no-return global_atomic_add_f32
//    into out[tgt] (output stays L2-resident; ~16 hits/row).
// ---------------------------------------------------------------------------
__global__ void __launch_bounds__(BLOCK)
hmp_scatter_kernel(const float* __restrict__ x,
                   const float* __restrict__ values,
                   const int*   __restrict__ src,
                   const int*   __restrict__ tgt,
                   float*       __restrict__ out,
                   int n_edges, int n_tiles) {
    // [wave][buffer][array: 0=val 1=src 2=tgt][edge-in-tile]
    __shared__ uint32_t smem[WAVES_PER_BLOCK][2][3][TILE];

    const int lane = threadIdx.x & 31;
    const int w    = threadIdx.x >> 5;
    const int gw   = blockIdx.x * WAVES_PER_BLOCK + w;
    const int n_waves = gridDim.x * WAVES_PER_BLOCK;

    // Raw LDS byte address of this wave's slice: flat-address low 32 bits
    const uint32_t lds_base = (uint32_t)(uintptr_t)(&smem[w][0][0][0]);

    const int half = lane >> 4;   // which of the 2 concurrent edges
    const int sub  = lane & 15;   // which float4 chunk of the 64-float row

    // Stage one tile of 32 edge records into LDS buffer `buf` (async, 3 ops).
    auto stage = [&](int tile, int buf) {
        int e = tile * TILE + lane;
        if (e >= n_edges) e = n_edges - 1;          // clamp (skipped later)
        const uint32_t goff = ((uint32_t)e) << 2;   // byte offset into arrays
        const uint32_t aV = lds_base + ((uint32_t)(((buf * 3 + 0) * TILE) + lane) << 2);
        const uint32_t aS = lds_base + ((uint32_t)(((buf * 3 + 1) * TILE) + lane) << 2);
        const uint32_t aT = lds_base + ((uint32_t)(((buf * 3 + 2) * TILE) + lane) << 2);
        asm volatile("global_load_async_to_lds_b32 %0, %1, %2 th:TH_LOAD_NT"
                     :: "v"(aV), "v"(goff), "s"(values) : "memory");
        asm volatile("global_load_async_to_lds_b32 %0, %1, %2 th:TH_LOAD_NT"
                     :: "v"(aS), "v"(goff), "s"(src) : "memory");
        asm volatile("global_load_async_to_lds_b32 %0, %1, %2 th:TH_LOAD_NT"
                     :: "v"(aT), "v"(goff), "s"(tgt) : "memory");
    };

    int tile = gw;
    if (tile >= n_tiles) return;

    int buf = 0;
    stage(tile, buf);                       // prime buffer 0

    for (; tile < n_tiles; tile += n_waves) {
        const int next = tile + n_waves;
        if (next < n_tiles) {
            stage(next, buf ^ 1);           // prefetch next tile (3 more ops)
            // 6 outstanding -> wait until only the 3 prefetch ops remain;
            // async loads complete in order, so current buffer is ready.
            asm volatile("s_wait_asynccnt 3" ::: "memory");
        } else {
            asm volatile("s_wait_asynccnt 0" ::: "memory");
        }

        const uint32_t* bV = &smem[w][buf][0][0];
        const uint32_t* bS = &smem[w][buf][1][0];
        const uint32_t* bT = &smem[w][buf][2][0];
        const int base_e = tile * TILE;

        #pragma unroll 4
        for (int k = 0; k < TILE / 2; ++k) {
            const int le = 2 * k + half;          // local edge for this half-wave
            const int e  = base_e + le;
            if (e < n_edges) {
                // LDS broadcast: all 16 lanes of the group read the same word.
                const float v = __uint_as_float(bV[le]);
                const int   s = (int)bS[le];
                const int   t = (int)bT[le];

                // Coalesced 256B gather per 16-lane group.
                const float4 xv =
                    reinterpret_cast<const float4*>(x + (size_t)s * D_FEAT)[sub];

                const float4 m = make_float4(v * xv.x, v * xv.y,
                                             v * xv.z, v * xv.w);

                float* o = out + (size_t)t * D_FEAT + (sub << 2);
                // No-return hardware f32 atomics (agent scope, relaxed).
                (void)__hip_atomic_fetch_add(o + 0, m.x, __ATOMIC_RELAXED,
                                             __HIP_MEMORY_SCOPE_AGENT);
                (void)__hip_atomic_fetch_add(o + 1, m.y, __ATOMIC_RELAXED,
                                             __HIP_MEMORY_SCOPE_AGENT);
                (void)__hip_atomic_fetch_add(o + 2, m.z, __ATOMIC_RELAXED,
                                             __HIP_MEMORY_SCOPE_AGENT);
                (void)__hip_atomic_fetch_add(o + 3, m.w, __ATOMIC_RELAXED,
                                             __HIP_MEMORY_SCOPE_AGENT);
            }
        }
        buf ^= 1;
    }
}

// ---------------------------------------------------------------------------
// Zero the output (harness poisons d_out with 0xAA; segment_sum semantics need
// zeros for nodes with no incoming edges).
// ---------------------------------------------------------------------------
__global__ void hmp_zero_kernel(float* __restrict__ out, int n4) {
    float4 z = make_float4(0.f, 0.f, 0.f, 0.f);
    for (int i = blockIdx.x * blockDim.x + threadIdx.x; i < n4;
         i += gridDim.x * blockDim.x) {
        reinterpret_cast<float4*>(out)[i] = z;
    }
}

// ---------------------------------------------------------------------------
// Launcher
// ---------------------------------------------------------------------------
extern "C" void kernel_launch(void* const* d_in, const int* in_sizes, int n_in,
                              void* d_out, int out_size, void* d_ws, size_t ws_size,
                              hipStream_t stream) {
    const float* x      = (const float*)d_in[0];
    const float* values = (const float*)d_in[1];
    const int*   src    = (const int*)d_in[2];
    const int*   tgt    = (const int*)d_in[3];
    // d_in[4] is n_target as a device scalar; out_size = n_target * D_FEAT.
    float* out = (float*)d_out;

    const int n_edges = in_sizes[1];

    // 1) zero output
    const int n4 = out_size / 4;
    int zb = (n4 + BLOCK - 1) / BLOCK;
    if (zb > 4096) zb = 4096;
    hmp_zero_kernel<<<zb, BLOCK, 0, stream>>>(out, n4);

    // 2) scatter-add
    const int n_tiles = (n_edges + TILE - 1) / TILE;
    int blocks = (n_tiles + WAVES_PER_BLOCK - 1) / WAVES_PER_BLOCK;
    if (blocks > MAX_BLOCKS) blocks = MAX_BLOCKS;   // ~4 tiles/wave grid-stride
    hmp_scatter_kernel<<<blocks, BLOCK, 0, stream>>>(x, values, src, tgt, out,
                                                     n_edges, n_tiles);
}